// ActorGCN_25726854103114
// MI455X (gfx1250) — compile-verified
//
#include <hip/hip_runtime.h>
#include <hip/hip_bf16.h>

// ---------------------------------------------------------------------------
// ActorGCN pipeline for MI455X (gfx1250, wave32).
//   h = x@W  (WMMA f32 16x16x4)  -> L2-resident (51.2MB)
//   deg/invsqrt, self-loop+bias init, edge scatter (f32 atomics in L2)
//   BN stats -> fused scale/shift, final linear(128->2)+relu+softmax+mask
// ---------------------------------------------------------------------------

#define F_IN 100
#define HD   128
#define OUTD 2
#define BN_EPS 1e-5f

typedef __attribute__((ext_vector_type(2))) float v2f;
typedef __attribute__((ext_vector_type(8))) float v8f;

// ---------------- GEMM: h = x @ W via V_WMMA_F32_16X16X4_F32 ----------------
// One wave owns a 16-row tile and sweeps all 8 column tiles (H=128):
// one float2 A-fragment load feeds 8 WMMAs per k-step. Block = 4 waves =
// 4 row tiles. Out-of-range rows are clamped (their outputs are never
// stored), so no EXEC-masked loads appear in the inner loop and EXEC stays
// all-ones for WMMA.
__global__ void __launch_bounds__(128)
gemm_wmma_f32(const float* __restrict__ x, const float* __restrict__ W,
              float* __restrict__ h, int Nn) {
  const int lane  = threadIdx.x & 31;
  const int wv    = threadIdx.x >> 5;            // 0..3 row tiles per block
  const int rbase = (blockIdx.x * 4 + wv) * 16;
  int row = rbase + (lane & 15);
  if (row >= Nn) row = Nn - 1;                   // clamp: safe, never stored
  const int khalf = (lane >> 4) << 1;            // 0 or 2 (A-frag K pair)
  const float* xr = x + (size_t)row * F_IN + khalf;  // 8B aligned
  const int col = lane & 15;

  v8f acc[8];
#pragma unroll
  for (int t = 0; t < 8; ++t) acc[t] = (v8f){};

  for (int k = 0; k < F_IN; k += 4) {
    v2f a = *(const v2f*)(xr + k);               // global_load_b64
    const float* wr = W + (size_t)(k + khalf) * HD + col;
#pragma unroll
    for (int t = 0; t < 8; ++t) {
      v2f b;
      b.x = wr[t * 16];          // W[k+khalf][col + 16t]
      b.y = wr[t * 16 + HD];     // W[k+khalf+1][col + 16t]
      acc[t] = __builtin_amdgcn_wmma_f32_16x16x4_f32(false, a, false, b,
                                                     (short)0, acc[t],
                                                     false, false);
    }
  }

  // C/D layout: VGPR v -> M = v + (lane>=16 ? 8 : 0), N = lane%16
  const int mrow0 = rbase + ((lane >> 4) << 3);
#pragma unroll
  for (int v = 0; v < 8; ++v) {
    const int m = mrow0 + v;
    if (m < Nn) {
      float* hr = h + (size_t)m * HD + col;
#pragma unroll
      for (int t = 0; t < 8; ++t) hr[t * 16] = acc[t][v];
    }
  }
}

// ---------------- degree (self loop => start at 1.0) ----------------
__global__ void init_deg(float* __restrict__ deg, int Nn) {
  int i = blockIdx.x * blockDim.x + threadIdx.x;
  if (i < Nn) deg[i] = 1.0f;
}

__global__ void count_deg(const int* __restrict__ dst, float* __restrict__ deg, int E) {
  int e = blockIdx.x * blockDim.x + threadIdx.x;
  if (e < E) atomicAdd(&deg[dst[e]], 1.0f);
}

__global__ void inv_sqrt_deg(const float* __restrict__ deg, float* __restrict__ isq, int Nn) {
  int i = blockIdx.x * blockDim.x + threadIdx.x;
  if (i < Nn) isq[i] = rsqrtf(deg[i]);
}

// ---------------- agg init: bias + self-loop contribution ----------------
// agg[i][:] = b[:] + h[i][:] * invsq[i]^2    (one thread per float4)
__global__ void self_bias_init(const float* __restrict__ h,
                               const float* __restrict__ isq,
                               const float* __restrict__ bias,
                               float* __restrict__ agg, int Nn) {
  int t = blockIdx.x * blockDim.x + threadIdx.x;    // Nn*32 threads
  int i = t >> 5, q = t & 31;
  if (i >= Nn) return;
  float w = isq[i]; w *= w;
  float4 v  = ((const float4*)(h + (size_t)i * HD))[q];
  float4 bb = ((const float4*)bias)[q];
  float4 r;
  r.x = bb.x + v.x * w; r.y = bb.y + v.y * w;
  r.z = bb.z + v.z * w; r.w = bb.w + v.w * w;
  ((float4*)(agg + (size_t)i * HD))[q] = r;
}

// ---------------- edge scatter: one wave per edge ----------------
// Lane q covers features 4q..4q+3; h[src] row (512B) gathered from L2,
// 4 fp32 atomic adds into agg[dst] (resolved at L2 on MI455X; h+agg=102MB
// both fit in the 192MB global L2).
__global__ void __launch_bounds__(256)
edge_scatter(const float* __restrict__ h, const float* __restrict__ isq,
             const int* __restrict__ src, const int* __restrict__ dst,
             float* __restrict__ agg, int E) {
  int wave = blockIdx.x * (blockDim.x >> 5) + (threadIdx.x >> 5);
  int lane = threadIdx.x & 31;
  if (wave >= E) return;
  int s = src[wave], d = dst[wave];
  float w = isq[s] * isq[d];
  float4 v = ((const float4*)(h + (size_t)s * HD))[lane];
  float* a = agg + (size_t)d * HD + lane * 4;
  atomicAdd(a + 0, v.x * w);
  atomicAdd(a + 1, v.y * w);
  atomicAdd(a + 2, v.z * w);
  atomicAdd(a + 3, v.w * w);
}

// ---------------- BatchNorm statistics ----------------
__global__ void bn_zero(float* __restrict__ sums) {
  int i = threadIdx.x;                 // 256 threads: sums[0..127]=sum, [128..255]=sumsq
  if (i < 2 * HD) sums[i] = 0.0f;
}

__global__ void __launch_bounds__(128)
bn_partial(const float* __restrict__ agg, float* __restrict__ sums, int Nn) {
  int c = threadIdx.x;                 // column 0..127
  float s = 0.0f, q = 0.0f;
  for (int r = blockIdx.x; r < Nn; r += gridDim.x) {
    float v = agg[(size_t)r * HD + c];
    s += v;
    q += v * v;
  }
  atomicAdd(&sums[c], s);
  atomicAdd(&sums[HD + c], q);
}

// fuse BN into per-column scale/shift: xn = agg*scale + shift
__global__ void bn_finalize(const float* __restrict__ sums,
                            const float* __restrict__ gamma,
                            const float* __restrict__ beta,
                            float* __restrict__ ss, int Nn) {
  int c = threadIdx.x;
  if (c >= HD) return;
  float invn = 1.0f / (float)Nn;
  float mean = sums[c] * invn;
  float var  = sums[HD + c] * invn - mean * mean;
  float sc   = gamma[c] * rsqrtf(var + BN_EPS);
  ss[c]      = sc;                    // scale
  ss[HD + c] = beta[c] - mean * sc;   // shift
}

// ---------------- final: linear(128->2) + relu + softmax + mask ----------------
// One wave per row; lane handles 4 strided columns, wave shuffle-reduce.
__global__ void __launch_bounds__(256)
final_head(const float* __restrict__ agg, const float* __restrict__ ss,
           const float* __restrict__ Wlin, const float* __restrict__ blin,
           const unsigned char* __restrict__ mask, float* __restrict__ out, int Nn) {
  int row  = blockIdx.x * (blockDim.x >> 5) + (threadIdx.x >> 5);
  int lane = threadIdx.x & 31;
  if (row >= Nn) return;
  float l0 = 0.0f, l1 = 0.0f;
#pragma unroll
  for (int j = 0; j < 4; ++j) {
    int c = lane + 32 * j;
    float v = agg[(size_t)row * HD + c] * ss[c] + ss[HD + c];
    l0 += v * Wlin[c * OUTD + 0];
    l1 += v * Wlin[c * OUTD + 1];
  }
#pragma unroll
  for (int off = 16; off > 0; off >>= 1) {
    l0 += __shfl_xor(l0, off, 32);
    l1 += __shfl_xor(l1, off, 32);
  }
  if (lane == 0) {
    l0 = fmaxf(l0 + blin[0], 0.0f);
    l1 = fmaxf(l1 + blin[1], 0.0f);
    float m  = fmaxf(l0, l1);
    float e0 = __expf(l0 - m), e1 = __expf(l1 - m);
    float inv = 1.0f / (e0 + e1);
    float keep = (mask[row] != 0) ? 1.0f : 0.0f;
    out[(size_t)row * OUTD + 0] = e0 * inv * keep;
    out[(size_t)row * OUTD + 1] = e1 * inv * keep;
  }
}

// ---------------------------------------------------------------------------
static inline size_t align256(size_t x) { return (x + 255) & ~(size_t)255; }

extern "C" void kernel_launch(void* const* d_in, const int* in_sizes, int n_in,
                              void* d_out, int out_size, void* d_ws, size_t ws_size,
                              hipStream_t stream) {
  const float* x     = (const float*)d_in[0];   // [N,10,10] -> [N,100]
  const float* W     = (const float*)d_in[1];   // [100,128]
  const float* bias  = (const float*)d_in[2];   // [128]
  const float* gamma = (const float*)d_in[3];   // [128]
  const float* beta  = (const float*)d_in[4];   // [128]
  const float* Wlin  = (const float*)d_in[5];   // [128,2]
  const float* blin  = (const float*)d_in[6];   // [2]
  const int*   eidx  = (const int*)d_in[7];     // [2,E]
  const unsigned char* mask = (const unsigned char*)d_in[8]; // bool [N]

  const int Nn = in_sizes[0] / F_IN;
  const int E  = in_sizes[7] / 2;
  const int* src = eidx;
  const int* dst = eidx + E;

  // workspace carve-up
  char* ws = (char*)d_ws;
  size_t off = 0;
  const size_t szNH = (size_t)Nn * HD * sizeof(float);
  float* h    = (float*)(ws + off); off = align256(off + szNH);
  float* agg  = (float*)(ws + off); off = align256(off + szNH);
  float* deg  = (float*)(ws + off); off = align256(off + (size_t)Nn * sizeof(float));
  float* isq  = (float*)(ws + off); off = align256(off + (size_t)Nn * sizeof(float));
  float* sums = (float*)(ws + off); off = align256(off + 2 * HD * sizeof(float));
  float* ss   = (float*)(ws + off); off = align256(off + 2 * HD * sizeof(float));
  (void)ws_size; (void)n_in; (void)out_size;

  float* out = (float*)d_out;

  // 1) h = x @ W (WMMA fp32): 4 row-tiles (64 rows) per block
  {
    int blocks = (Nn + 63) / 64;
    gemm_wmma_f32<<<blocks, 128, 0, stream>>>(x, W, h, Nn);
  }
  // 2) degree + inv-sqrt
  {
    int b = 256;
    init_deg<<<(Nn + b - 1) / b, b, 0, stream>>>(deg, Nn);
    count_deg<<<(E + b - 1) / b, b, 0, stream>>>(dst, deg, E);
    inv_sqrt_deg<<<(Nn + b - 1) / b, b, 0, stream>>>(deg, isq, Nn);
  }
  // 3) agg = bias + self-loop term
  {
    long long t = (long long)Nn * 32;
    self_bias_init<<<(unsigned)((t + 255) / 256), 256, 0, stream>>>(h, isq, bias, agg, Nn);
  }
  // 4) edge scatter (wave per edge)
  {
    int wavesPerBlk = 8;
    edge_scatter<<<(E + wavesPerBlk - 1) / wavesPerBlk, 256, 0, stream>>>(
        h, isq, src, dst, agg, E);
  }
  // 5) batch-norm stats -> fused scale/shift
  bn_zero<<<1, 256, 0, stream>>>(sums);
  bn_partial<<<1024, 128, 0, stream>>>(agg, sums, Nn);
  bn_finalize<<<1, 128, 0, stream>>>(sums, gamma, beta, ss, Nn);
  // 6) final head
  {
    int rowsPerBlk = 8;
    final_head<<<(Nn + rowsPerBlk - 1) / rowsPerBlk, 256, 0, stream>>>(
        agg, ss, Wlin, blin, mask, out, Nn);
  }
}